// LiftSplatShoot_44092134261500
// MI455X (gfx1250) — compile-verified
//
#include <hip/hip_runtime.h>
#include <hip/hip_bf16.h>

// ---- problem constants (from reference setup) ----
#define BDIM   4
#define NDIM   6
#define BN     24          // B*N
#define CIN    80
#define HDIM   16
#define WDIM   44
#define HW     704         // H*W
#define DBINS  59
#define COUT   83          // 3 + CIN
#define DSTEP  (0.9f / 58.0f)   // linspace(0.1, 1.0, 59) step

// ---- workspace layout (floats) ----
#define WS_MT  0                     // per-bn: M[9] (=R*K^-1 row-major), t[3]  -> 24*12
#define WS_DL  (BN * 12)             // d_last: 24*704
#define WS_WR  (WS_DL + BN * HW)     // wray:   24*704*3

typedef __attribute__((ext_vector_type(2))) float v2f;
typedef __attribute__((ext_vector_type(8))) float v8f;

// ------------------------------------------------------------------
// Kernel 1: per-(b,n) prep.  M = R * inv(K[:3,:3]), t = extr[:3,3].
// 24 lanes of one wave do 24 independent 3x3 inverses + 3x3 matmuls.
// ------------------------------------------------------------------
__global__ void k_prep(const float* __restrict__ intr,
                       const float* __restrict__ extr,
                       float* __restrict__ ws) {
    int i = threadIdx.x;
    if (i >= BN) return;
    const float* k = intr + i * 16;
    float a = k[0], b = k[1], c = k[2];
    float d = k[4], e = k[5], f = k[6];
    float g = k[8], h = k[9], ii = k[10];
    float A0 = e * ii - f * h, A1 = c * h - b * ii, A2 = b * f - c * e;
    float A3 = f * g - d * ii, A4 = a * ii - c * g, A5 = c * d - a * f;
    float A6 = d * h - e * g,  A7 = b * g - a * h,  A8 = a * e - b * d;
    float det = a * A0 + b * A3 + c * A6;
    float inv = 1.0f / det;
    float Ki[9] = {A0 * inv, A1 * inv, A2 * inv,
                   A3 * inv, A4 * inv, A5 * inv,
                   A6 * inv, A7 * inv, A8 * inv};
    const float* ex = extr + i * 16;
    float* o = ws + WS_MT + i * 12;
    for (int r = 0; r < 3; ++r) {
        for (int cc = 0; cc < 3; ++cc) {
            o[r * 3 + cc] = ex[r * 4 + 0] * Ki[0 * 3 + cc]
                          + ex[r * 4 + 1] * Ki[1 * 3 + cc]
                          + ex[r * 4 + 2] * Ki[2 * 3 + cc];
        }
        o[9 + r] = ex[r * 4 + 3];
    }
}

// ------------------------------------------------------------------
// Kernel 2: d_last[bn][p] = softmax over p (704 elems) of channel 79.
// One block per bn.
// ------------------------------------------------------------------
__global__ void k_softmax(const float* __restrict__ feats,
                          float* __restrict__ ws) {
    __shared__ float red[256];
    const int bn = blockIdx.x;
    const int t = threadIdx.x;
    const float* x = feats + ((long long)bn * CIN + (CIN - 1)) * HW;

    float m = -3.4e38f;
    for (int p = t; p < HW; p += 256) m = fmaxf(m, x[p]);
    red[t] = m; __syncthreads();
    for (int s = 128; s > 0; s >>= 1) {
        if (t < s) red[t] = fmaxf(red[t], red[t + s]);
        __syncthreads();
    }
    const float mx = red[0];
    __syncthreads();

    float sum = 0.0f;
    for (int p = t; p < HW; p += 256) sum += __expf(x[p] - mx);
    red[t] = sum; __syncthreads();
    for (int s = 128; s > 0; s >>= 1) {
        if (t < s) red[t] += red[t + s];
        __syncthreads();
    }
    const float rinv = 1.0f / red[0];

    for (int p = t; p < HW; p += 256)
        ws[WS_DL + bn * HW + p] = __expf(x[p] - mx) * rinv;
}

// ------------------------------------------------------------------
// Kernel 3: wray[bn][p][0:3] = d_last[bn][p] * (M · [x,y,1]) via
// V_WMMA_F32_16X16X4_F32.  One wave transforms 16 pixels:
//   A (16x4): rows 0..2 = M | 0-col, rows 3..15 = 0
//   B (4x16): column n = [x_n, y_n, 1, 0]
//   D (16x16): rows 0..2 = transformed rays.
// A layout: lanes 0-15 -> {K0,K1}, lanes 16-31 -> {K2,K3}. B mirrors.
// 24 bn * 44 groups = 1056 waves exactly (no partial waves, EXEC all-1).
// ------------------------------------------------------------------
__global__ void k_rays_wmma(float* __restrict__ ws) {
    const int tid  = blockIdx.x * blockDim.x + threadIdx.x;
    const int wid  = tid >> 5;           // global wave id, 0..1055
    const int lane = tid & 31;
    const int bn   = wid / 44;
    const int g    = wid - bn * 44;
    const int p0   = g * 16;

    const int half = lane >> 4;          // 0: K=0/1 ; 1: K=2/3
    const int l    = lane & 15;

    // A operand: M rows (row-major 3x3 in ws), zero-padded to 16x4
    const float* M = ws + WS_MT + bn * 12;
    v2f a;
    a.x = (l < 3) ? ((half == 0) ? M[l * 3 + 0] : M[l * 3 + 2]) : 0.0f;
    a.y = (l < 3 && half == 0) ? M[l * 3 + 1] : 0.0f;

    // B operand: 16 pixel columns [x, y, 1, 0]
    const int p = p0 + l;
    const int h = p / WDIM;
    const int w = p - h * WDIM;
    v2f bm;
    bm.x = (half == 0) ? (float)w : 1.0f;   // K=0 row: x   | K=2 row: 1
    bm.y = (half == 0) ? (float)h : 0.0f;   // K=1 row: y   | K=3 row: 0

    v8f acc = {};
    acc = __builtin_amdgcn_wmma_f32_16x16x4_f32(
        /*neg_a=*/false, a, /*neg_b=*/false, bm,
        /*c_mod=*/(short)0, acc, /*reuse_a=*/false, /*reuse_b=*/false);

    // Rows M=0..2 live in acc[0..2] on lanes 0..15 (N = lane)
    if (lane < 16) {
        const int pp = p0 + lane;
        const float dl = ws[WS_DL + bn * HW + pp];
        float* wr = ws + WS_WR + (bn * HW + pp) * 3;
        wr[0] = acc[0] * dl;
        wr[1] = acc[1] * dl;
        wr[2] = acc[2] * dl;
    }
}

// ------------------------------------------------------------------
// Kernel 4: the bandwidth kernel. One wave per output row of 59 floats.
// rows = BN * 83 * 704 = 1,402,368;  out[row*59 + d]:
//   c <  3: wray[bn][p][c] * depth[d] + t[c]
//   c >= 3: feats[bn][c-3][p]  (broadcast over depth)
// Lanes cover d = lane and d = lane + 32 -> one contiguous 236B burst.
// Output (331 MB) is write-once streaming > L2 (192 MB): use
// NON-TEMPORAL stores so the hot read set (feats, ws) stays in L2.
// ------------------------------------------------------------------
__global__ void k_write(const float* __restrict__ feats,
                        const float* __restrict__ ws,
                        float* __restrict__ out) {
    const int lane = threadIdx.x & 31;
    const int row  = blockIdx.x * 8 + (threadIdx.x >> 5);   // < 1,402,368

    const int bn  = row / (COUT * HW);
    const int rem = row - bn * (COUT * HW);
    const int c   = rem / HW;
    const int p   = rem - c * HW;

    float* o = out + (long long)row * DBINS;

    if (c < 3) {
        const float wr = ws[WS_WR + (bn * HW + p) * 3 + c];
        const float tc = ws[WS_MT + bn * 12 + 9 + c];
        __builtin_nontemporal_store(
            fmaf(wr, 0.1f + (float)lane * DSTEP, tc), o + lane);
        if (lane < DBINS - 32)
            __builtin_nontemporal_store(
                fmaf(wr, 0.1f + (float)(lane + 32) * DSTEP, tc), o + lane + 32);
    } else {
        const float v = feats[((long long)bn * CIN + (c - 3)) * HW + p];
        __builtin_nontemporal_store(v, o + lane);
        if (lane < DBINS - 32)
            __builtin_nontemporal_store(v, o + lane + 32);
    }
}

// ------------------------------------------------------------------
extern "C" void kernel_launch(void* const* d_in, const int* in_sizes, int n_in,
                              void* d_out, int out_size, void* d_ws, size_t ws_size,
                              hipStream_t stream) {
    const float* feats = (const float*)d_in[0];   // (24, 80, 16, 44) f32
    const float* intr  = (const float*)d_in[1];   // (4, 6, 4, 4)  f32
    const float* extr  = (const float*)d_in[2];   // (4, 6, 4, 4)  f32
    float* out = (float*)d_out;                   // (4,6,83,16,44,59) f32
    float* ws  = (float*)d_ws;                    // ~272 KB used

    // 1) M = R * K^-1 and t (24 lanes, one wave)
    k_prep<<<1, 32, 0, stream>>>(intr, extr, ws);

    // 2) spatial softmax of last channel -> d_last (one block per bn)
    k_softmax<<<BN, 256, 0, stream>>>(feats, ws);

    // 3) WMMA ray transform: 1056 waves = 132 blocks * 8 waves
    k_rays_wmma<<<132, 256, 0, stream>>>(ws);

    // 4) stream out 331 MB with NT stores: 1,402,368 rows, 8 waves/block
    const int rows = BN * COUT * HW;      // 1,402,368
    k_write<<<rows / 8, 256, 0, stream>>>(feats, ws, out);
}